// Module_1_69655779607239
// MI455X (gfx1250) — compile-verified
//
#include <hip/hip_runtime.h>

// Problem constants (from reference): B=128, T=512, N=200, H=128
#define BB 128
#define TT 512
#define NN 200
#define HH 128
#define NP 208              // N padded to 13*16
#define NTILE 13            // NP/16 row tiles
#define NTILES (NTILE*NTILE) // 169 tiles for NPxNP outputs
#define GRAM_BPB 22         // ceil(169/8) blocks per batch for gram

typedef __attribute__((ext_vector_type(2))) float v2f;
typedef __attribute__((ext_vector_type(8))) float v8f;

// ---------------------------------------------------------------------------
// fp32 WMMA step: D(16x16) += A(16x4) * B(4x16), full fp32 precision.
// A layout (ISA 7.12.2, 32-bit A 16x4): lane 0-15 -> M=lane, VGPR0=K0,VGPR1=K1;
// lane 16-31 -> M=lane-16, VGPR0=K2, VGPR1=K3.  (K = 2*half + vgpr)
// ---------------------------------------------------------------------------
__device__ __forceinline__ v8f wmma4(v8f acc, float a0, float a1, float b0, float b1) {
    v2f av = {a0, a1};
    v2f bv = {b0, b1};
    return __builtin_amdgcn_wmma_f32_16x16x4_f32(false, av, false, bv, (short)0, acc,
                                                 false, false);
}
__device__ __forceinline__ v8f wmma4v(v8f acc, v2f av, v2f bv) {
    return __builtin_amdgcn_wmma_f32_16x16x4_f32(false, av, false, bv, (short)0, acc,
                                                 false, false);
}

// ---------------------------------------------------------------------------
// 1) Per-(b,n) column mean of X[b, :, n] over T
// ---------------------------------------------------------------------------
__global__ void means_kernel(const float* __restrict__ X, float* __restrict__ mu) {
    const int b = blockIdx.x;
    const int c = threadIdx.x;
    if (c >= NN) return;
    const float* Xb = X + (size_t)b * TT * NN;
    float s = 0.f;
    for (int t = 0; t < TT; ++t) s += Xb[(size_t)t * NN + c];
    mu[b * NN + c] = s * (1.0f / (float)TT);
}

// ---------------------------------------------------------------------------
// 2) Batched gram: G[b,i,j] = sum_t (X[b,t,i]-mu_i)(X[b,t,j]-mu_j)
//    One 16x16 output tile per wave, WMMA over T=512.
//    Out-of-range lanes (i>=200) are handled by clamped addresses +
//    multiplicative zero masks -> all loads unconditional (no EXEC branches).
//    The 1/(T-1) factor cancels in the correlation normalization -> skipped.
// ---------------------------------------------------------------------------
__global__ void __launch_bounds__(256)
gram_kernel(const float* __restrict__ X, const float* __restrict__ mu,
            float* __restrict__ G) {
    const int b    = blockIdx.x / GRAM_BPB;
    const int tb   = blockIdx.x % GRAM_BPB;
    const int wave = threadIdx.x >> 5;
    const int tile = tb * 8 + wave;
    if (tile >= NTILES) return;
    const int ti = tile / NTILE, tj = tile % NTILE;
    const int lane = threadIdx.x & 31;
    const int l = lane & 15, h = lane >> 4;
    const int i = ti * 16 + l;   // A-fragment row
    const int j = tj * 16 + l;   // B-fragment col
    const int iS = (i < NN) ? i : 0;       // clamped (always-valid) addresses
    const int jS = (j < NN) ? j : 0;
    const float wA = (i < NN) ? 1.f : 0.f; // multiplicative pad masks
    const float wB = (j < NN) ? 1.f : 0.f;
    const float* Xb = X + (size_t)b * TT * NN;
    const float muA = mu[b * NN + iS];
    const float muB = mu[b * NN + jS];
    v8f acc0 = {}, acc1 = {};
#pragma unroll 4
    for (int k0 = 0; k0 < TT; k0 += 8) {
        const int k = k0 + 2 * h;
        const float a0 = (Xb[(size_t)k * NN + iS] - muA) * wA;
        const float a1 = (Xb[(size_t)(k + 1) * NN + iS] - muA) * wA;
        const float b0 = (Xb[(size_t)k * NN + jS] - muB) * wB;
        const float b1 = (Xb[(size_t)(k + 1) * NN + jS] - muB) * wB;
        const int k2 = k + 4;
        const float c0 = (Xb[(size_t)k2 * NN + iS] - muA) * wA;
        const float c1 = (Xb[(size_t)(k2 + 1) * NN + iS] - muA) * wA;
        const float d0 = (Xb[(size_t)k2 * NN + jS] - muB) * wB;
        const float d1 = (Xb[(size_t)(k2 + 1) * NN + jS] - muB) * wB;
        acc0 = wmma4(acc0, a0, a1, b0, b1);
        acc1 = wmma4(acc1, c0, c1, d0, d1);
    }
    const v8f acc = acc0 + acc1;
    float* Gb = G + (size_t)b * NP * NP;
#pragma unroll
    for (int r = 0; r < 8; ++r) {
        const int row = ti * 16 + r + 8 * h;
        Gb[(size_t)row * NP + tj * 16 + l] = acc[r];
    }
}

// 3a) dd[b,n] = sqrt(G[b,n,n])
__global__ void diag_kernel(const float* __restrict__ G, float* __restrict__ dd) {
    const int idx = blockIdx.x * 256 + threadIdx.x;
    if (idx >= BB * NP) return;
    const int b = idx / NP, n = idx % NP;
    dd[idx] = sqrtf(G[(size_t)b * NP * NP + (size_t)n * NP + n]);
}

// 3b) In place: G <- |clip(G/(d_n d_m))| with reference NaN/inf semantics:
//     NaN -> 0 first, then clip (so +-inf -> +-1).
__global__ void vabs_kernel(float* __restrict__ G, const float* __restrict__ dd) {
    const size_t idx = (size_t)blockIdx.x * 256 + threadIdx.x;
    const size_t total = (size_t)BB * NP * NP;
    if (idx >= total) return;
    const size_t per = (size_t)NP * NP;
    const int b = (int)(idx / per);
    const int rem = (int)(idx % per);
    const int n = rem / NP, m = rem % NP;
    const float denom = dd[b * NP + n] * dd[b * NP + m];
    float fc = G[idx] / denom;
    if (fc != fc) fc = 0.f;                 // NaN (0/0 on padded rows) -> 0
    fc = fminf(1.f, fmaxf(-1.f, fc));       // clip handles +-inf -> +-1
    G[idx] = fabsf(fc);                      // node feature v = |fc|
}

// ---------------------------------------------------------------------------
// 4) U[b] = Vabs[b] (NPxNP, ld NP) @ W (NNxHH).  K = 200 (25 dual-WMMA steps).
//    Block = 8 waves = 8 column tiles (HH=128); grid = B * NTILE row tiles.
//    A-side pairs are contiguous & 8B aligned -> float2 loads.
// ---------------------------------------------------------------------------
__global__ void __launch_bounds__(256)
gemm_aw_kernel(const float* __restrict__ A, const float* __restrict__ W,
               float* __restrict__ Cout) {
    const int b  = blockIdx.x / NTILE;
    const int ti = blockIdx.x % NTILE;
    const int tj = threadIdx.x >> 5;
    const int lane = threadIdx.x & 31;
    const int l = lane & 15, h = lane >> 4;
    const int i = ti * 16 + l;
    const int col = tj * 16 + l;
    const float* Ab = A + (size_t)b * NP * NP + (size_t)i * NP;
    v8f acc0 = {}, acc1 = {};
#pragma unroll 5
    for (int k0 = 0; k0 < NN; k0 += 8) {
        const int k = k0 + 2 * h;
        const v2f a01 = *(const v2f*)(Ab + k);
        const v2f a23 = *(const v2f*)(Ab + k + 4);
        const v2f b01 = {W[(size_t)k * HH + col], W[(size_t)(k + 1) * HH + col]};
        const v2f b23 = {W[(size_t)(k + 4) * HH + col], W[(size_t)(k + 5) * HH + col]};
        acc0 = wmma4v(acc0, a01, b01);
        acc1 = wmma4v(acc1, a23, b23);
    }
    const v8f acc = acc0 + acc1;
    float* Cb = Cout + (size_t)b * NP * HH;
#pragma unroll
    for (int r = 0; r < 8; ++r) {
        const int row = ti * 16 + r + 8 * h;
        Cb[(size_t)row * HH + col] = acc[r];
    }
}

// ---------------------------------------------------------------------------
// 5) Z[b] = mask(Vabs[b]) @ U[b] + eps*U[b] + bias.  mask = (v != 0) -> {0,1}
// ---------------------------------------------------------------------------
__global__ void __launch_bounds__(256)
gemm_mask_kernel(const float* __restrict__ V, const float* __restrict__ U,
                 const float* __restrict__ epsPtr, const float* __restrict__ bias,
                 float* __restrict__ Zout) {
    const int b  = blockIdx.x / NTILE;
    const int ti = blockIdx.x % NTILE;
    const int tj = threadIdx.x >> 5;
    const int lane = threadIdx.x & 31;
    const int l = lane & 15, h = lane >> 4;
    const int i = ti * 16 + l;
    const int col = tj * 16 + l;
    const float* Vb = V + (size_t)b * NP * NP + (size_t)i * NP;
    const float* Ub = U + (size_t)b * NP * HH;
    v8f acc0 = {}, acc1 = {};
#pragma unroll 5
    for (int k0 = 0; k0 < NN; k0 += 8) {
        const int k = k0 + 2 * h;
        const v2f v01 = *(const v2f*)(Vb + k);
        const v2f v23 = *(const v2f*)(Vb + k + 4);
        const v2f a01 = {(v01[0] != 0.f) ? 1.f : 0.f, (v01[1] != 0.f) ? 1.f : 0.f};
        const v2f a23 = {(v23[0] != 0.f) ? 1.f : 0.f, (v23[1] != 0.f) ? 1.f : 0.f};
        const v2f b01 = {Ub[(size_t)k * HH + col], Ub[(size_t)(k + 1) * HH + col]};
        const v2f b23 = {Ub[(size_t)(k + 4) * HH + col], Ub[(size_t)(k + 5) * HH + col]};
        acc0 = wmma4v(acc0, a01, b01);
        acc1 = wmma4v(acc1, a23, b23);
    }
    const v8f acc = acc0 + acc1;
    const float eps = epsPtr[0];
    const float bc  = bias[col];
    float* Zb = Zout + (size_t)b * NP * HH;
#pragma unroll
    for (int r = 0; r < 8; ++r) {
        const int row = ti * 16 + r + 8 * h;
        Zb[(size_t)row * HH + col] = acc[r] + eps * Ub[(size_t)row * HH + col] + bc;
    }
}

// ---------------------------------------------------------------------------
// 6) Z[b] = Hin[b] (NPxHH) @ W (HHxHH) [+ bias].  K = 128 (16 dual-WMMA steps).
// ---------------------------------------------------------------------------
__global__ void __launch_bounds__(256)
gemm_hw_kernel(const float* __restrict__ Hin, const float* __restrict__ W,
               const float* __restrict__ bias, float* __restrict__ Zout) {
    const int b  = blockIdx.x / NTILE;
    const int ti = blockIdx.x % NTILE;
    const int tj = threadIdx.x >> 5;
    const int lane = threadIdx.x & 31;
    const int l = lane & 15, h = lane >> 4;
    const int i = ti * 16 + l;
    const int col = tj * 16 + l;
    const float* Ab = Hin + (size_t)b * NP * HH + (size_t)i * HH;
    v8f acc0 = {}, acc1 = {};
#pragma unroll 4
    for (int k0 = 0; k0 < HH; k0 += 8) {
        const int k = k0 + 2 * h;
        const v2f a01 = *(const v2f*)(Ab + k);
        const v2f a23 = *(const v2f*)(Ab + k + 4);
        const v2f b01 = {W[(size_t)k * HH + col], W[(size_t)(k + 1) * HH + col]};
        const v2f b23 = {W[(size_t)(k + 4) * HH + col], W[(size_t)(k + 5) * HH + col]};
        acc0 = wmma4v(acc0, a01, b01);
        acc1 = wmma4v(acc1, a23, b23);
    }
    const v8f acc = acc0 + acc1;
    const float bc = bias ? bias[col] : 0.f;
    float* Zb = Zout + (size_t)b * NP * HH;
#pragma unroll
    for (int r = 0; r < 8; ++r) {
        const int row = ti * 16 + r + 8 * h;
        Zb[(size_t)row * HH + col] = acc[r] + bc;
    }
}

// ---------------------------------------------------------------------------
// 7) Training-mode BN stats over the 25600 valid rows (n<200), per channel.
//    Deterministic: fixed per-thread order + fixed tree reduction.
//    Emits fused affine: scale = g*rsqrt(var+eps), shift = be - mean*scale.
// ---------------------------------------------------------------------------
__global__ void bn_stats_kernel(const float* __restrict__ Z, const float* __restrict__ g,
                                const float* __restrict__ be,
                                float* __restrict__ scale, float* __restrict__ shift) {
    const int c = blockIdx.x;
    const int tid = threadIdx.x;
    float s = 0.f, q = 0.f;
    const int R = BB * NN;  // 25600 rows
    for (int r = tid; r < R; r += 256) {
        const int b = r / NN, n = r % NN;
        const float v = Z[(size_t)b * NP * HH + (size_t)n * HH + c];
        s += v;
        q += v * v;
    }
    __shared__ float ss[256], sq[256];
    ss[tid] = s; sq[tid] = q;
    __syncthreads();
    for (int st = 128; st > 0; st >>= 1) {
        if (tid < st) { ss[tid] += ss[tid + st]; sq[tid] += sq[tid + st]; }
        __syncthreads();
    }
    if (tid == 0) {
        const float m   = ss[0] / (float)R;
        const float var = sq[0] / (float)R - m * m;   // biased variance
        const float sc  = g[c] * rsqrtf(var + 1e-5f);
        scale[c] = sc;
        shift[c] = be[c] - m * sc;
    }
}

// 8) H = relu(Z*scale + shift); padded rows (n>=200) forced to zero.
__global__ void bn_relu_kernel(const float* __restrict__ Z, const float* __restrict__ scale,
                               const float* __restrict__ shift, float* __restrict__ Hout) {
    const size_t idx = (size_t)blockIdx.x * 256 + threadIdx.x;
    const size_t total = (size_t)BB * NP * HH;
    if (idx >= total) return;
    const int c = (int)(idx % HH);
    const int n = (int)((idx / HH) % NP);
    float v = 0.f;
    if (n < NN) {
        v = Z[idx] * scale[c] + shift[c];
        v = v > 0.f ? v : 0.f;
    }
    Hout[idx] = v;
}

// 9) Final BN+ReLU writing dense [B, N, H] output (outer relu is idempotent).
__global__ void bn_relu_out_kernel(const float* __restrict__ Z, const float* __restrict__ scale,
                                   const float* __restrict__ shift, float* __restrict__ out) {
    const size_t idx = (size_t)blockIdx.x * 256 + threadIdx.x;
    const size_t total = (size_t)BB * NN * HH;
    if (idx >= total) return;
    const int c = (int)(idx % HH);
    const size_t row = idx / HH;          // 0 .. B*NN-1
    const int b = (int)(row / NN);
    const int n = (int)(row % NN);
    float v = Z[(size_t)b * NP * HH + (size_t)n * HH + c] * scale[c] + shift[c];
    out[idx] = v > 0.f ? v : 0.f;
}

// ---------------------------------------------------------------------------
extern "C" void kernel_launch(void* const* d_in, const int* in_sizes, int n_in,
                              void* d_out, int out_size, void* d_ws, size_t ws_size,
                              hipStream_t stream) {
    const float* X      = (const float*)d_in[0];
    const float* g1_eps = (const float*)d_in[1];
    const float* g1_w1  = (const float*)d_in[2];
    const float* g1_b1  = (const float*)d_in[3];
    const float* g1_g1  = (const float*)d_in[4];
    const float* g1_be1 = (const float*)d_in[5];
    const float* g1_w2  = (const float*)d_in[6];
    const float* g1_b2  = (const float*)d_in[7];
    const float* g1_g2  = (const float*)d_in[8];
    const float* g1_be2 = (const float*)d_in[9];
    const float* g2_eps = (const float*)d_in[10];
    const float* g2_w1  = (const float*)d_in[11];
    const float* g2_b1  = (const float*)d_in[12];
    const float* g2_g1  = (const float*)d_in[13];
    const float* g2_be1 = (const float*)d_in[14];
    const float* g2_w2  = (const float*)d_in[15];
    const float* g2_b2  = (const float*)d_in[16];
    const float* g2_g2  = (const float*)d_in[17];
    const float* g2_be2 = (const float*)d_in[18];
    float* out = (float*)d_out;

    // Workspace partitioning (floats)
    float* ws    = (float*)d_ws;
    float* mu    = ws;                                   // B*NN
    float* dd    = mu    + (size_t)BB * NN;              // B*NP
    float* G     = dd    + (size_t)BB * NP;              // B*NP*NP  (gram -> |fc| in place)
    float* U     = G     + (size_t)BB * NP * NP;         // B*NP*HH
    float* Z     = U     + (size_t)BB * NP * HH;         // B*NP*HH
    float* Hb    = Z     + (size_t)BB * NP * HH;         // B*NP*HH
    float* scale = Hb    + (size_t)BB * NP * HH;         // HH
    float* shift = scale + HH;                           // HH

    const int gemmGrid = BB * NTILE;                     // 1664 blocks, 8 waves each
    const int ewBlocks = (BB * NP * HH) / 256;           // 13312
    const int vaBlocks = (int)(((size_t)BB * NP * NP) / 256);  // 21632

    // --- correlation ---
    means_kernel<<<BB, 256, 0, stream>>>(X, mu);
    gram_kernel<<<BB * GRAM_BPB, 256, 0, stream>>>(X, mu, G);
    diag_kernel<<<(BB * NP + 255) / 256, 256, 0, stream>>>(G, dd);
    vabs_kernel<<<vaBlocks, 256, 0, stream>>>(G, dd);    // G now holds |fc|

    // --- GIN layer 1:  z1 = mask@(v@W1) + eps1*(v@W1) + b1 ---
    gemm_aw_kernel<<<gemmGrid, 256, 0, stream>>>(G, g1_w1, U);
    gemm_mask_kernel<<<gemmGrid, 256, 0, stream>>>(G, U, g1_eps, g1_b1, Z);
    bn_stats_kernel<<<HH, 256, 0, stream>>>(Z, g1_g1, g1_be1, scale, shift);
    bn_relu_kernel<<<ewBlocks, 256, 0, stream>>>(Z, scale, shift, Hb);
    gemm_hw_kernel<<<gemmGrid, 256, 0, stream>>>(Hb, g1_w2, g1_b2, Z);
    bn_stats_kernel<<<HH, 256, 0, stream>>>(Z, g1_g2, g1_be2, scale, shift);
    bn_relu_kernel<<<ewBlocks, 256, 0, stream>>>(Z, scale, shift, Hb);  // Hb = x1

    // --- GIN layer 2:  z3 = mask@(x1@W1') + eps2*(x1@W1') + b1' ---
    gemm_hw_kernel<<<gemmGrid, 256, 0, stream>>>(Hb, g2_w1, nullptr, U);
    gemm_mask_kernel<<<gemmGrid, 256, 0, stream>>>(G, U, g2_eps, g2_b1, Z);
    bn_stats_kernel<<<HH, 256, 0, stream>>>(Z, g2_g1, g2_be1, scale, shift);
    bn_relu_kernel<<<ewBlocks, 256, 0, stream>>>(Z, scale, shift, Hb);
    gemm_hw_kernel<<<gemmGrid, 256, 0, stream>>>(Hb, g2_w2, g2_b2, Z);
    bn_stats_kernel<<<HH, 256, 0, stream>>>(Z, g2_g2, g2_be2, scale, shift);
    bn_relu_out_kernel<<<(BB * NN * HH) / 256, 256, 0, stream>>>(Z, scale, shift, out);

    (void)in_sizes; (void)n_in; (void)out_size; (void)ws_size;
}